// HashConv2d_73882027426042
// MI455X (gfx1250) — compile-verified
//
#include <hip/hip_runtime.h>
#include <hip/hip_bf16.h>

// ---------------------------------------------------------------------------
// HashedNets 3x3 conv (C=256->256, 32x64x64, fp32) as implicit GEMM on the
// CDNA5 matrix cores: V_WMMA_F32_16X16X4_F32 (fp32 preserves reference math).
//
// Pass 1: materialize hashed virtual weight (2.36 MB) into d_ws, pre-swizzled
//         into the WMMA A-fragment pair layout  Wt[r][ci>>1][co][ci&1].
//         Hash: n_blocks = 1024 (power of two) -> pure VALU.
// Pass 2: per workgroup: one (n, h) output row x 128 c_out. 8 waves, each
//         16 c_out x 64 px (4 x v8f fp32 accumulators).
//         Input staged via GLOBAL_LOAD_ASYNC_TO_LDS_B32 into double-buffered
//         LDS (2 x 16 ci-pairs x 3 rows x 66 cols x 2 = 50.7 KB), ci-pair
//         interleaved so each B fragment is one aligned ds_load_b64.
//         Pipeline: issue asyncs for chunk c+1 -> 288 WMMAs on chunk c ->
//         s_wait_asynccnt 0 -> barrier.
// ---------------------------------------------------------------------------

typedef __attribute__((ext_vector_type(2))) float v2f;
typedef __attribute__((ext_vector_type(8))) float v8f;

#define C_OUT     256
#define C_IN      256
#define HW        64
#define CI_CHUNK  32                 // ci per LDS stage
#define NCHUNK    (C_IN / CI_CHUNK)  // 8
#define LDS_ROW   66                 // 64 + 2 halo columns

// ---------------- Pass 1: hashed-weight materialization --------------------
__global__ __launch_bounds__(256)
void hashconv_build_wt(const float* __restrict__ pool, float* __restrict__ wt) {
    int v = blockIdx.x * blockDim.x + threadIdx.x;   // 0 .. 589823
    if (v >= C_OUT * C_IN * 9) return;
    unsigned uv  = (unsigned)v;
    unsigned blk = uv >> 6;
    unsigned off = uv & 63u;
    unsigned h   = blk * 2654435761u;                // Knuth, seed = 0
    unsigned idx = ((h & 1023u) << 6) | off;         // n_blocks = 1024 (pow2)
    float w = pool[idx];
    // v = co*2304 + ci*9 + r   (reference reshape order)
    int r  = v % 9;
    int t  = v / 9;
    int ci = t & 255;
    int co = t >> 8;
    wt[((((r << 7) + (ci >> 1)) << 8) + co) * 2 + (ci & 1)] = w;
}

// ---------------- async chunk copy: global -> LDS (interior cells) ---------
// Interior = 32 ci x 3 rows x 64 cols; halo cols (0,65) and OOB edge rows
// are pre-zeroed once and never touched by the async copies.
__device__ __forceinline__
void hashconv_async_chunk(float* __restrict__ dstbase,   // lds buf base
                          const float* __restrict__ x,
                          int n, int h, int ci0, int tid) {
    int ci  = (tid >= 192) ? 1 : 0;
    int rem = tid - ci * 192;                 // rem in [0,192): row*64 + ww
    #pragma unroll 4
    for (int k = 0; k < 24; ++k) {            // 24 * 256 = 6144 cells
        int row = rem >> 6;                   // 0..2
        int ww  = rem & 63;                   // 0..63
        int hh  = h - 1 + row;
        if ((unsigned)hh < (unsigned)HW) {    // EXEC-masks OOB edge rows
            const float* gp = x + (((n * C_IN + ci0 + ci) * HW + hh) << 6) + ww;
            float* lp = dstbase
                      + ((((ci >> 1) * 3 + row) * LDS_ROW + (ww + 1)) << 1)
                      + (ci & 1);
            unsigned loff = (unsigned)(unsigned long long)lp;  // low 32b = LDS addr
            asm volatile("global_load_async_to_lds_b32 %0, %1, off"
                         :: "v"(loff), "v"(gp) : "memory");
        }
        ci  += 1;
        rem += 64;
        if (rem >= 192) { rem -= 192; ci += 1; }
    }
}

// ---------------- Pass 2: implicit-GEMM conv on WMMA ------------------------
__global__ __launch_bounds__(256, 1)
void hashconv_wmma(const float* __restrict__ x,
                   const float* __restrict__ wt,
                   const float* __restrict__ bias,
                   const float* __restrict__ lfactor,
                   float* __restrict__ out) {
    // [buf][ci/2][row][col][ci&1]  -> B fragment = one aligned ds_load_b64
    __shared__ float lds_x[2][CI_CHUNK / 2][3][LDS_ROW][2];   // 50688 B

    int bx     = blockIdx.x;                 // (n*64 + h)*2 + mhalf
    int mhalf  = bx & 1;
    int nh     = bx >> 1;
    int n      = nh >> 6;
    int h      = nh & 63;
    int Mbase  = mhalf << 7;

    int tid  = threadIdx.x;
    int lane = tid & 31;
    int wv   = tid >> 5;                     // 8 waves
    int Mw   = Mbase + (wv << 4);            // this wave's 16 c_out rows
    int nloc = lane & 15;
    int hi   = lane >> 4;                    // lanes 16-31 carry K+2 (A/B layout)

    // -------- one-time halo zeroing for BOTH buffers -----------------------
    for (int i = tid; i < 2 * CI_CHUNK * 3 * 2; i += 256) {   // 384 cells
        int buf = i / 192;
        int r2  = i - buf * 192;
        int ci  = r2 / 6;
        int rr  = (r2 - ci * 6) >> 1;
        int cc  = (r2 & 1) ? (LDS_ROW - 1) : 0;
        lds_x[buf][ci >> 1][rr][cc][ci & 1] = 0.0f;
    }
    if (h == 0 || h == 63) {                 // OOB edge row: zero whole row
        int er = (h == 0) ? 0 : 2;
        for (int i = tid; i < 2 * CI_CHUNK * LDS_ROW; i += 256) {  // 4224
            int buf = i / (CI_CHUNK * LDS_ROW);
            int r2  = i - buf * (CI_CHUNK * LDS_ROW);
            int ci  = r2 / LDS_ROW;
            int cc  = r2 - ci * LDS_ROW;
            lds_x[buf][ci >> 1][er][cc][ci & 1] = 0.0f;
        }
    }

    v8f acc0 = {}, acc1 = {}, acc2 = {}, acc3 = {};

    // -------- prologue: async-stage chunk 0 --------------------------------
    hashconv_async_chunk(&lds_x[0][0][0][0][0], x, n, h, 0, tid);
    asm volatile("s_wait_asynccnt 0x0" ::: "memory");
    __syncthreads();

    for (int c = 0; c < NCHUNK; ++c) {
        int cur = c & 1;
        if (c + 1 < NCHUNK)   // overlap: issue next chunk's async copies now
            hashconv_async_chunk(&lds_x[cur ^ 1][0][0][0][0], x, n, h,
                                 (c + 1) * CI_CHUNK, tid);

        const float* lbase = &lds_x[cur][0][0][0][0];

        for (int r = 0; r < 9; ++r) {
            int kh = r / 3;
            int kw = r - kh * 3;
            // A-fragment base for this tap: Wt[r][ci/2][co][2]
            const float* wrow = wt + (((r << 7) + (c << 4) + hi) << 9)
                                   + ((Mw + nloc) << 1);
            #pragma unroll
            for (int ks = 0; ks < CI_CHUNK / 4; ++ks) {    // 8 k-steps of 4
                // A: lane m=nloc loads pair {W[k0+koff][co], W[k0+koff+1][co]}
                v2f a = *(const v2f*)(wrow + (ks << 10));

                int p = (ks << 1) + hi;                    // ci-pair index
                const float* brow = lbase
                    + (((p * 3 + kh) * LDS_ROW + kw + nloc) << 1);

                #pragma unroll
                for (int nsub = 0; nsub < 4; ++nsub) {
                    v2f b = *(const v2f*)(brow + (nsub << 5));  // ds_load_b64
                    v8f* accp = (nsub == 0) ? &acc0 : (nsub == 1) ? &acc1
                              : (nsub == 2) ? &acc2 : &acc3;
                    *accp = __builtin_amdgcn_wmma_f32_16x16x4_f32(
                        /*neg_a=*/false, a, /*neg_b=*/false, b,
                        /*c_mod=*/(short)0, *accp,
                        /*reuse_a=*/false, /*reuse_b=*/false);
                }
            }
        }
        asm volatile("s_wait_asynccnt 0x0" ::: "memory");  // next buf landed
        __syncthreads();
    }

    // ---------------- epilogue: (acc + bias) * layer_factor ----------------
    float lf = lfactor[0];
    #pragma unroll
    for (int nsub = 0; nsub < 4; ++nsub) {
        const v8f& a = (nsub == 0) ? acc0 : (nsub == 1) ? acc1
                     : (nsub == 2) ? acc2 : acc3;
        int wcol = (nsub << 4) + nloc;
        #pragma unroll
        for (int e = 0; e < 8; ++e) {
            int co = Mw + (hi << 3) + e;     // C/D layout: vgpr e -> M=e / M=e+8
            float v = (a[e] + bias[co]) * lf;
            out[(((n * C_OUT + co) * HW + h) << 6) + wcol] = v;
        }
    }
}

// ---------------------------------------------------------------------------
extern "C" void kernel_launch(void* const* d_in, const int* in_sizes, int n_in,
                              void* d_out, int out_size, void* d_ws, size_t ws_size,
                              hipStream_t stream) {
    const float* x    = (const float*)d_in[0];   // (32,256,64,64)
    const float* pool = (const float*)d_in[1];   // (65536,)
    const float* bias = (const float*)d_in[2];   // (256,)
    const float* lf   = (const float*)d_in[3];   // (1,)
    float* out = (float*)d_out;                  // (32,256,64,64)
    float* wt  = (float*)d_ws;                   // 589824 floats = 2.36 MB

    hipLaunchKernelGGL(hashconv_build_wt, dim3(2304), dim3(256), 0, stream,
                       pool, wt);
    hipLaunchKernelGGL(hashconv_wmma, dim3(4096), dim3(256), 0, stream,
                       x, wt, bias, lf, out);
}